// Classifier_16913581212020
// MI455X (gfx1250) — compile-verified
//
#include <hip/hip_runtime.h>

// CDNA5 / gfx1250 fused gather->concat->fc1->relu->fc2 using native f32 WMMA.
// Round 2: 64 rows per workgroup (4 M-tiles x 4 N-tiles per wave) to cut
// L2 weight re-read traffic 4x and raise WMMA:VMEM ratio to 4:1.
//
// Shapes: B=512, N=1024, M=128, OUT_DIM=MOLVEC_DIM=256, IN_DIM=512.
// Output: [B*M, 512] f32.

typedef __attribute__((ext_vector_type(2))) float v2f;
typedef __attribute__((ext_vector_type(8))) float v8f;

static constexpr int kB   = 512;
static constexpr int kN   = 1024;
static constexpr int kM   = 128;
static constexpr int kOut = 256;            // OUT_DIM
static constexpr int kMol = 256;            // MOLVEC_DIM
static constexpr int kIn  = 512;            // IN_DIM
static constexpr int kCat = kOut + kMol;    // 512 = fc1 fan-in
static constexpr int kLdsStride = kCat + 4; // pad: spreads A-frag lanes across banks, keeps 8B align
static constexpr int kRows = 64;            // rows per workgroup (4 WMMA M-tiles)
static constexpr int kMT   = 4;             // M-tiles per wave
static constexpr int kNT   = 4;             // N-tiles per wave (4*16 = 64 cols)

__global__ __launch_bounds__(256, 1) void mlp_fused_f32_wmma(
    const float* __restrict__ X,       // [B, N, OUT]
    const float* __restrict__ molvec,  // [B, MOL]
    const int*   __restrict__ idxM,    // [B, M]
    const float* __restrict__ W1,      // [IN, CAT] row-major (torch layout)
    const float* __restrict__ b1,      // [IN]
    const float* __restrict__ W2,      // [IN, IN]
    const float* __restrict__ b2,      // [IN]
    float* __restrict__ out)           // [B*M, IN]
{
    __shared__ float tileA[kRows * kLdsStride];   // 64 x 516 f32 = 132 KB (CDNA5: <=320KB/WG)

    const int tid  = threadIdx.x;
    const int wave = tid >> 5;       // 0..7
    const int lane = tid & 31;
    const int half = lane >> 4;      // 0: lanes 0-15, 1: lanes 16-31
    const int ln16 = lane & 15;
    const int rowBase = blockIdx.x * kRows;   // global row (b*M + m) base

    // ---------------- Phase 0: gather + concat into LDS ----------------
    // 64 rows x 512 cols, 256 threads: 4 threads/row, 128 contiguous cols each.
    // Each thread's span lies entirely in X (c0<256) or molvec (c0>=256).
    {
        const int r  = tid >> 2;             // row in tile, 0..63
        const int c0 = (tid & 3) * 128;      // col start: 0,128,256,384
        const int g  = rowBase + r;
        const int b  = g >> 7;               // / kM
        const int m  = g & (kM - 1);
        const int id = idxM[b * kM + m];     // 0..N-1
        const float* __restrict__ src =
            (c0 < kOut) ? X + ((size_t)b * kN + (size_t)id) * kOut + c0
                        : molvec + (size_t)b * kMol + (c0 - kOut);
        float4* __restrict__ dst = (float4*)(tileA + r * kLdsStride + c0);
#pragma unroll
        for (int j = 0; j < 32; ++j)
            dst[j] = ((const float4*)src)[j];
    }
    __syncthreads();

    const int nbase = wave * 64;   // this wave's 64 output columns

    // ---------------- Phase 1: h = relu(A @ W1^T + b1) ----------------
    // A-frag (16x4 f32): lanes 0-15 hold K=k..k+1, lanes 16-31 K=k+2..k+3, row = lane%16.
    // B-frag (4x16):     lanes hold N = lane%16; VGPR0/1 = K rows (k+2*half, +1).
    v8f acc[kMT][kNT];
#pragma unroll
    for (int mt = 0; mt < kMT; ++mt)
#pragma unroll
        for (int nt = 0; nt < kNT; ++nt)
            acc[mt][nt] = (v8f){};

    {
        const float* __restrict__ arow[kMT];
#pragma unroll
        for (int mt = 0; mt < kMT; ++mt)
            arow[mt] = tileA + (mt * 16 + ln16) * kLdsStride + 2 * half;
        const float* __restrict__ wrow[kNT];
#pragma unroll
        for (int nt = 0; nt < kNT; ++nt)
            wrow[nt] = W1 + (size_t)(nbase + nt * 16 + ln16) * kCat + 2 * half;

#pragma unroll 4
        for (int k = 0; k < kCat; k += 4) {
            v2f a[kMT], f[kNT];
#pragma unroll
            for (int mt = 0; mt < kMT; ++mt) a[mt] = *(const v2f*)(arow[mt] + k);
#pragma unroll
            for (int nt = 0; nt < kNT; ++nt) f[nt] = *(const v2f*)(wrow[nt] + k);
#pragma unroll
            for (int mt = 0; mt < kMT; ++mt)
#pragma unroll
                for (int nt = 0; nt < kNT; ++nt)
                    acc[mt][nt] = __builtin_amdgcn_wmma_f32_16x16x4_f32(
                        false, a[mt], false, f[nt], (short)0, acc[mt][nt], false, false);
        }
    }
    __syncthreads();   // all waves done reading tileA

    // bias + relu, write h over tileA (64 x 512)
    {
#pragma unroll
        for (int nt = 0; nt < kNT; ++nt) {
            const float bb = b1[nbase + nt * 16 + ln16];
#pragma unroll
            for (int mt = 0; mt < kMT; ++mt) {
#pragma unroll
                for (int r = 0; r < 8; ++r) {
                    const int mrow = mt * 16 + r + 8 * half;  // C/D layout: VGPR r -> M=r(+8 hi lanes)
                    tileA[mrow * kLdsStride + nbase + nt * 16 + ln16] =
                        fmaxf(acc[mt][nt][r] + bb, 0.0f);
                }
            }
        }
    }
    __syncthreads();

    // ---------------- Phase 2: pred = h @ W2^T + b2 ----------------
#pragma unroll
    for (int mt = 0; mt < kMT; ++mt)
#pragma unroll
        for (int nt = 0; nt < kNT; ++nt)
            acc[mt][nt] = (v8f){};

    {
        const float* __restrict__ arow[kMT];
#pragma unroll
        for (int mt = 0; mt < kMT; ++mt)
            arow[mt] = tileA + (mt * 16 + ln16) * kLdsStride + 2 * half;
        const float* __restrict__ wrow[kNT];
#pragma unroll
        for (int nt = 0; nt < kNT; ++nt)
            wrow[nt] = W2 + (size_t)(nbase + nt * 16 + ln16) * kIn + 2 * half;

#pragma unroll 4
        for (int k = 0; k < kIn; k += 4) {
            v2f a[kMT], f[kNT];
#pragma unroll
            for (int mt = 0; mt < kMT; ++mt) a[mt] = *(const v2f*)(arow[mt] + k);
#pragma unroll
            for (int nt = 0; nt < kNT; ++nt) f[nt] = *(const v2f*)(wrow[nt] + k);
#pragma unroll
            for (int mt = 0; mt < kMT; ++mt)
#pragma unroll
                for (int nt = 0; nt < kNT; ++nt)
                    acc[mt][nt] = __builtin_amdgcn_wmma_f32_16x16x4_f32(
                        false, a[mt], false, f[nt], (short)0, acc[mt][nt], false, false);
        }
    }

    // bias + coalesced store
    {
#pragma unroll
        for (int nt = 0; nt < kNT; ++nt) {
            const float bb = b2[nbase + nt * 16 + ln16];
#pragma unroll
            for (int mt = 0; mt < kMT; ++mt) {
#pragma unroll
                for (int r = 0; r < 8; ++r) {
                    const int mrow = mt * 16 + r + 8 * half;
                    out[(size_t)(rowBase + mrow) * kIn + nbase + nt * 16 + ln16] =
                        acc[mt][nt][r] + bb;
                }
            }
        }
    }
}

extern "C" void kernel_launch(void* const* d_in, const int* in_sizes, int n_in,
                              void* d_out, int out_size, void* d_ws, size_t ws_size,
                              hipStream_t stream) {
    (void)in_sizes; (void)n_in; (void)out_size; (void)d_ws; (void)ws_size;
    const float* X      = (const float*)d_in[0];
    const float* molvec = (const float*)d_in[1];
    const int*   idxM   = (const int*)d_in[2];
    const float* W1     = (const float*)d_in[3];
    const float* b1     = (const float*)d_in[4];
    const float* W2     = (const float*)d_in[5];
    const float* b2     = (const float*)d_in[6];
    float*       out    = (float*)d_out;

    const int totalRows = kB * kM;            // 65536
    dim3 grid(totalRows / kRows);             // 1024 workgroups, 64 rows each
    dim3 block(256);                          // 8 wave32s
    hipLaunchKernelGGL(mlp_fused_f32_wmma, grid, block, 0, stream,
                       X, molvec, idxM, W1, b1, W2, b2, out);
}